// GCNConv_15779709845755
// MI455X (gfx1250) — compile-verified
//
#include <hip/hip_runtime.h>

typedef float v2f __attribute__((ext_vector_type(2)));
typedef float v8f __attribute__((ext_vector_type(8)));

#define GN 512   // graph size N
#define CIO 128  // C_IN == C_OUT
#define RS  640  // row stride in floats (N + C_IN)

// ---------------------------------------------------------------------------
// Kernel 1: copy A -> out[:, :, :512], compute deg row sums, dis = rsqrt(deg)
// One wave (32 lanes) per (b, n) row; 8 rows per 256-thread block.
// ---------------------------------------------------------------------------
__global__ __launch_bounds__(256) void gcn_deg_copy(const float* __restrict__ X,
                                                    float* __restrict__ out,
                                                    float* __restrict__ dis) {
    int wave = threadIdx.x >> 5;
    int lane = threadIdx.x & 31;
    long r = (long)blockIdx.x * 8 + wave;   // global row id: b*512 + n
    const float* src = X + r * RS;
    float* dst = out + r * RS;
    float sum = 0.f;
#pragma unroll
    for (int i = 0; i < 4; ++i) {
        float4 v = *(const float4*)(src + i * 128 + lane * 4);
        *(float4*)(dst + i * 128 + lane * 4) = v;
        sum += v.x + v.y + v.z + v.w;
    }
#pragma unroll
    for (int off = 16; off > 0; off >>= 1)
        sum += __shfl_xor(sum, off, 32);
    if (lane == 0)
        dis[r] = (sum > 0.f) ? rsqrtf(sum) : 0.f;
}

// ---------------------------------------------------------------------------
// Kernel 2: per (b, 16-row tile): h = dis_n*(A @ (dis.*feat) + dis_n*feat_n),
// then out = relu(h @ W). 8 waves; wave w owns 16-wide column strip.
// ---------------------------------------------------------------------------
__global__ __launch_bounds__(256) void gcn_gemm(const float* __restrict__ X,
                                                const float* __restrict__ W,
                                                const float* __restrict__ dis,
                                                float* __restrict__ out) {
    __shared__ float sA[16 * 516];   // 16x512 A tile, pad 516 (conflict-free)
    __shared__ float sDis[GN];       // dis row for this batch
    __shared__ float sH[16 * 132];   // 16x128 h tile, pad 132

    int b    = blockIdx.x >> 5;
    int n0   = (blockIdx.x & 31) << 4;
    int tid  = threadIdx.x;
    int wave = tid >> 5;
    int lane = tid & 31;
    int l16  = lane & 15;
    int kh   = lane >> 4;    // 0 or 1 (K-half select per WMMA layout)

    const float* Xb = X + (size_t)b * GN * RS;

    // Stage A tile (16 x 512) into LDS, coalesced float4 loads
    for (int idx = tid; idx < 16 * 128; idx += 256) {
        int row = idx >> 7;
        int c4  = (idx & 127) << 2;
        float4 v = *(const float4*)(Xb + (size_t)(n0 + row) * RS + c4);
        *(float4*)(&sA[row * 516 + c4]) = v;
    }
    for (int i = tid; i < GN; i += 256)
        sDis[i] = dis[(size_t)b * GN + i];
    __syncthreads();

    // ---- GEMM1: acc(16x16) += A(16x4) * featS(4x16), K = 512 ----
    // A operand layout (f32 16x4): lanes 0-15 = M rows; VGPR0 = K {0 | 2},
    // VGPR1 = K {1 | 3} by lane half -> read float2 at (row, k0 + 2*kh).
    const float* fbase = Xb + GN + (wave << 4) + l16;  // feat(m,f) = fbase[m*RS]
    const float* aBase = &sA[l16 * 516 + (kh << 1)];
    v8f acc = {};
#pragma unroll 4
    for (int k0 = 0; k0 < GN; k0 += 4) {
        v2f a = *(const v2f*)(aBase + k0);
        int m0 = k0 + (kh << 1);
        v2f bv;
        bv.x = fbase[(size_t)m0 * RS] * sDis[m0];
        bv.y = fbase[(size_t)(m0 + 1) * RS] * sDis[m0 + 1];
        acc = __builtin_amdgcn_wmma_f32_16x16x4_f32(false, a, false, bv,
                                                    (short)0, acc, false, false);
    }

    // h = dis_n * acc + dis_n^2 * feat(n, f); C/D layout: VGPR j -> M = j + 8*kh
#pragma unroll
    for (int j = 0; j < 8; ++j) {
        int rl = j + (kh << 3);
        float dn = sDis[n0 + rl];
        float fv = fbase[(size_t)(n0 + rl) * RS];
        sH[rl * 132 + (wave << 4) + l16] = dn * (acc[j] + dn * fv);
    }
    __syncthreads();

    // ---- GEMM2: out(16x16) += h(16x4) * W(4x16), K = 128 ----
    const float* wbase = W + (wave << 4) + l16;        // W(k,o) = wbase[k*128]
    const float* hBase = &sH[l16 * 132 + (kh << 1)];
    v8f acc2 = {};
#pragma unroll 4
    for (int k0 = 0; k0 < CIO; k0 += 4) {
        v2f a = *(const v2f*)(hBase + k0);
        int m0 = k0 + (kh << 1);
        v2f bv;
        bv.x = wbase[(size_t)m0 * CIO];
        bv.y = wbase[(size_t)(m0 + 1) * CIO];
        acc2 = __builtin_amdgcn_wmma_f32_16x16x4_f32(false, a, false, bv,
                                                     (short)0, acc2, false, false);
    }

    // relu + store into out[:, :, 512:640]
    float* obase = out + (size_t)b * GN * RS + GN + (wave << 4) + l16;
#pragma unroll
    for (int j = 0; j < 8; ++j) {
        int rl = j + (kh << 3);
        obase[(size_t)(n0 + rl) * RS] = fmaxf(acc2[j], 0.f);
    }
}

// ---------------------------------------------------------------------------
extern "C" void kernel_launch(void* const* d_in, const int* in_sizes, int n_in,
                              void* d_out, int out_size, void* d_ws, size_t ws_size,
                              hipStream_t stream) {
    const float* X = (const float*)d_in[0];   // (128, 512, 640) fp32
    const float* W = (const float*)d_in[1];   // (128, 128) fp32
    float* out = (float*)d_out;               // (128, 512, 640) fp32
    float* dis = (float*)d_ws;                // 128*512 floats = 256 KB scratch

    const int B = 128;
    // Kernel 1: 8 rows per block, B*N rows total
    gcn_deg_copy<<<B * GN / 8, 256, 0, stream>>>(X, out, dis);
    // Kernel 2: one block per (batch, 16-row tile)
    gcn_gemm<<<B * (GN / 16), 256, 0, stream>>>(X, W, dis, out);
}